// LapLoss_57191784514139
// MI455X (gfx1250) — compile-verified
//
#include <hip/hip_runtime.h>
#include <math.h>

// LapLoss, MI455X (gfx1250, wave32).
// Strategy: lap pyramid is linear -> build pyramid of (logit - target) once.
// One fused kernel per level (64x64 lap tile / 32x32 down tile per WG):
//   LDS-stage 72x72 reflect-padded tile -> WMMA f32 16x16x4 banded matmuls for
//   (5-tap blur + 2x decimate) in both axes -> down interior, VALU halo ring,
//   fused upsample + |laplacian| partial reduction. Deterministic reductions.
// Bandwidth-bound: ~200 MB total HBM traffic -> ~9us floor @ 23.3 TB/s.
// Workspace requirement: 5,592,400 floats (~22.4 MB).

typedef __attribute__((ext_vector_type(2))) float v2f;
typedef __attribute__((ext_vector_type(8))) float v8f;

#define THREADS 256

// binomial 5-tap /16, zero outside [0,4]
__device__ __forceinline__ float bandw(int d) {
  float w = 0.0f;
  if (d == 0 || d == 4) w = 0.0625f;
  else if (d == 1 || d == 3) w = 0.25f;
  else if (d == 2) w = 0.375f;
  return w;
}

__launch_bounds__(THREADS)
__global__ void lap_level_kernel(const float* __restrict__ inA,
                                 const float* __restrict__ inB,  // non-null only at level 0
                                 const int H, const int W,
                                 float* __restrict__ outDown,
                                 float* __restrict__ partial) {
  __shared__ float inT[72][73];   // 64x64 tile + halo 4, reflect-padded
  __shared__ float hbd[80][33];   // h-blur + col-decimate; rows 0..67 valid
                                  // (padded to 5 full 16-row WMMA blocks so
                                  //  D-fragment stores need no EXEC masking)
  __shared__ float dwn[34][35];   // 32x32 down interior + halo ring 1
  __shared__ float red[THREADS];

  const int tid = threadIdx.x;
  const int ntx = W >> 6;
  const int tpi = (H >> 6) * ntx;
  const int img = blockIdx.x / tpi;
  const int trem = blockIdx.x % tpi;
  const int y0 = (trem / ntx) << 6;
  const int x0 = (trem % ntx) << 6;
  const size_t base = (size_t)img * H * W;
  const bool twoIn = (inB != nullptr);

  // ---- Phase 1: stage reflect-padded 72x72 tile (difference image at level 0)
  for (int p = tid; p < 72 * 72; p += THREADS) {
    const int ly = p / 72, lx = p % 72;
    int gy = y0 - 4 + ly;
    gy = (gy < 0) ? -gy : ((gy >= H) ? (2 * H - 2 - gy) : gy);
    int gx = x0 - 4 + lx;
    gx = (gx < 0) ? -gx : ((gx >= W) ? (2 * W - 2 - gx) : gx);
    const size_t gidx = base + (size_t)gy * W + gx;
    float val = inA[gidx];
    if (twoIn) val -= inB[gidx];
    inT[ly][lx] = val;
  }
  __syncthreads();

  const int lane = tid & 31;
  const int half = lane >> 4;   // lanes 0-15 hold K{0,1}, lanes 16-31 hold K{2,3}
  const int mr = lane & 15;
  const int wave = tid >> 5;

#if __has_builtin(__builtin_amdgcn_wmma_f32_16x16x4_f32)
  // ---- Phase 2 (WMMA): horizontal blur + column decimation.
  // hbd[r][n] = sum_t g[t] * inT[r+2][2n+t+2]  as  A(16xK data) x B(Kx16 band), K=36.
  // 80x32 output (rows >=68 are don't-care) -> 5 row-blocks x 2 col-blocks over 8 waves.
  for (int blk = wave; blk < 10; blk += 8) {
    const int rb = blk >> 1, nb = blk & 1;
    v8f acc = {0.f, 0.f, 0.f, 0.f, 0.f, 0.f, 0.f, 0.f};
    int ir = (rb << 4) + mr + 2;
    if (ir > 71) ir = 71;  // rows >=68 read clamped garbage; never consumed
#pragma unroll
    for (int c = 0; c < 9; ++c) {
      const int kb = 4 * c + 2 * half;
      v2f a, b;
      a.x = inT[ir][(nb << 5) + kb + 2];
      a.y = inT[ir][(nb << 5) + kb + 3];
      b.x = bandw(kb - 2 * mr);        // B[k][n] = g[k-2n]
      b.y = bandw(kb + 1 - 2 * mr);
      acc = __builtin_amdgcn_wmma_f32_16x16x4_f32(false, a, false, b,
                                                  (short)0, acc, false, false);
    }
#pragma unroll
    for (int e = 0; e < 8; ++e)        // D: row = 16*rb + e + 8*half (unguarded)
      hbd[(rb << 4) + (half << 3) + e][(nb << 4) + mr] = acc[e];
  }
  __syncthreads();
  // ---- Phase 3 (WMMA): vertical blur + row decimation.
  // dwn[m][n] = sum_t g[t] * hbd[2m+t][n]  as  A(16xK band) x B(Kx16 data), K=36.
  if (wave < 4) {
    const int mb = wave >> 1, nb = wave & 1;
    v8f acc = {0.f, 0.f, 0.f, 0.f, 0.f, 0.f, 0.f, 0.f};
#pragma unroll
    for (int c = 0; c < 9; ++c) {
      const int kb = 4 * c + 2 * half;
      v2f a, b;
      a.x = bandw(kb - 2 * mr);        // A[m][k] = g[k-2m]
      a.y = bandw(kb + 1 - 2 * mr);
      const int rr = (mb << 5) + kb;   // K window shifts by 32 per M-block
      b.x = hbd[rr][(nb << 4) + mr];
      b.y = hbd[rr + 1][(nb << 4) + mr];
      acc = __builtin_amdgcn_wmma_f32_16x16x4_f32(false, a, false, b,
                                                  (short)0, acc, false, false);
    }
#pragma unroll
    for (int e = 0; e < 8; ++e)
      dwn[(mb << 4) + (half << 3) + e + 1][(nb << 4) + mr + 1] = acc[e];
  }
#else
  // ---- VALU fallback (same math) — used only if the f32 WMMA builtin is absent
  for (int p = tid; p < 68 * 32; p += THREADS) {
    const int r = p >> 5, n = p & 31;
    float s = 0.f;
#pragma unroll
    for (int t = 0; t < 5; ++t) s += bandw(t) * inT[r + 2][2 * n + t + 2];
    hbd[r][n] = s;
  }
  __syncthreads();
  for (int p = tid; p < 32 * 32; p += THREADS) {
    const int m = p >> 5, n = p & 31;
    float s = 0.f;
#pragma unroll
    for (int t = 0; t < 5; ++t) s += bandw(t) * hbd[2 * m + t][n];
    dwn[m + 1][n + 1] = s;
  }
#endif
  __syncthreads();

  // ---- Phase 4: down halo ring (132 cells) directly from inT.
  // Equals reference's zero-interleave reflect semantics because the kernel is
  // symmetric and the pad is reflect (filtered[-2]==filtered[2], etc.).
  if (tid < 132) {
    int r, c2;
    if (tid < 34)       { r = 0;         c2 = tid; }
    else if (tid < 68)  { r = 33;        c2 = tid - 34; }
    else if (tid < 100) { r = tid - 67;  c2 = 0; }
    else                { r = tid - 99;  c2 = 33; }
    const int dr = r - 1, dc = c2 - 1;
    float s = 0.f;
#pragma unroll
    for (int i = 0; i < 5; ++i) {
      float rs = 0.f;
#pragma unroll
      for (int j = 0; j < 5; ++j)
        rs += bandw(j) * inT[2 * dr + i + 2][2 * dc + j + 2];
      s += bandw(i) * rs;
    }
    dwn[r][c2] = s;
  }
  __syncthreads();

  // ---- Phase 5a: store interior 32x32 down block
  const int Hh = H >> 1, Wh = W >> 1;
  const size_t dbase = (size_t)img * Hh * Wh;
  for (int p = tid; p < 1024; p += THREADS) {
    const int m = p >> 5, n = p & 31;
    outDown[dbase + (size_t)((y0 >> 1) + m) * Wh + (x0 >> 1) + n] = dwn[m + 1][n + 1];
  }

  // ---- Phase 5b: fused upsample + laplacian + |.| accumulation.
  // up = (2gv) x (2gh):  even phase taps {1,6,1}/8, odd phase taps {1,1}/2.
  float acc = 0.f;
  for (int p = tid; p < 4096; p += THREADS) {
    const int ty = p >> 6, tx = p & 63;
    const int hy = ty >> 1, hx = tx >> 1;
    float wy0, wy1, wy2; int ry, nry;
    if (ty & 1) { wy0 = 0.5f;   wy1 = 0.5f;  wy2 = 0.f;    ry = hy + 1; nry = 2; }
    else        { wy0 = 0.125f; wy1 = 0.75f; wy2 = 0.125f; ry = hy;     nry = 3; }
    float wx0, wx1, wx2; int cx, ncx;
    if (tx & 1) { wx0 = 0.5f;   wx1 = 0.5f;  wx2 = 0.f;    cx = hx + 1; ncx = 2; }
    else        { wx0 = 0.125f; wx1 = 0.75f; wx2 = 0.125f; cx = hx;     ncx = 3; }
    float up = 0.f;
    for (int a2 = 0; a2 < nry; ++a2) {
      const float wv = (a2 == 0) ? wy0 : ((a2 == 1) ? wy1 : wy2);
      float rs = wx0 * dwn[ry + a2][cx] + wx1 * dwn[ry + a2][cx + 1];
      if (ncx == 3) rs += wx2 * dwn[ry + a2][cx + 2];
      up += wv * rs;
    }
    acc += fabsf(inT[ty + 4][tx + 4] - up);
  }

  // ---- Phase 6: deterministic block tree reduction
  red[tid] = acc;
  __syncthreads();
  for (int s = THREADS / 2; s > 0; s >>= 1) {
    if (tid < s) red[tid] += red[tid + s];
    __syncthreads();
  }
  if (tid == 0) partial[blockIdx.x] = red[0];
}

__launch_bounds__(256)
__global__ void lap_final_kernel(const float* __restrict__ partial,
                                 float* __restrict__ out) {
  __shared__ float red[256];
  const int tid = threadIdx.x;
  const int offs[6] = {0, 4096, 5120, 5376, 5440, 5456};
  const float scale[5] = {1.0f / 16777216.0f,   // 2^0 / (16*1024*1024)
                          2.0f / 4194304.0f,    // 2^1 / (16*512*512)
                          4.0f / 1048576.0f,    // 2^2 / (16*256*256)
                          8.0f / 262144.0f,     // 2^3 / (16*128*128)
                          16.0f / 65536.0f};    // 2^4 / (16*64*64)
  float acc = 0.f;
  for (int l = 0; l < 5; ++l)
    for (int i = offs[l] + tid; i < offs[l + 1]; i += 256)
      acc += scale[l] * partial[i];
  red[tid] = acc;
  __syncthreads();
  for (int s = 128; s > 0; s >>= 1) {
    if (tid < s) red[tid] += red[tid + s];
    __syncthreads();
  }
  if (tid == 0) out[0] = red[0];
}

extern "C" void kernel_launch(void* const* d_in, const int* in_sizes, int n_in,
                              void* d_out, int out_size, void* d_ws, size_t ws_size,
                              hipStream_t stream) {
  (void)in_sizes; (void)n_in; (void)out_size; (void)ws_size;
  const float* logit  = (const float*)d_in[0];
  const float* target = (const float*)d_in[1];
  float* ws = (float*)d_ws;
  float* d1 = ws;               // 16*512*512  = 4194304 f
  float* d2 = d1 + 4194304;     // 16*256*256  = 1048576 f
  float* d3 = d2 + 1048576;     // 16*128*128  =  262144 f
  float* d4 = d3 + 262144;      // 16*64*64    =   65536 f
  float* d5 = d4 + 65536;       // 16*32*32    =   16384 f (level-4 down, unused)
  float* P  = d5 + 16384;       // 5456 partial sums

  lap_level_kernel<<<dim3(4096), dim3(THREADS), 0, stream>>>(logit, target, 1024, 1024, d1, P);
  lap_level_kernel<<<dim3(1024), dim3(THREADS), 0, stream>>>(d1, nullptr, 512, 512, d2, P + 4096);
  lap_level_kernel<<<dim3(256),  dim3(THREADS), 0, stream>>>(d2, nullptr, 256, 256, d3, P + 5120);
  lap_level_kernel<<<dim3(64),   dim3(THREADS), 0, stream>>>(d3, nullptr, 128, 128, d4, P + 5376);
  lap_level_kernel<<<dim3(16),   dim3(THREADS), 0, stream>>>(d4, nullptr, 64, 64, d5, P + 5440);
  lap_final_kernel<<<dim3(1), dim3(256), 0, stream>>>(P, (float*)d_out);
}